// Model_77309411328338
// MI455X (gfx1250) — compile-verified
//
#include <hip/hip_runtime.h>
#include <hip/hip_bf16.h>
#include <math.h>

// ---------------------------------------------------------------------------
// AMU (Associative Memory Unit) — MI455X / gfx1250
// B=64, T=512, Din=128, Dhid=512, Dout=128, NUM_SLOTS=8, SLOT_DIM=3, RATE=0.1
//
// Strategy: hoist all batched GEMMs out of the serial scans and run them on
// the fp32 WMMA path (V_WMMA_F32_16X16X4_F32). Scans keep the tiny [8,3]
// memory state per batch element in LDS/registers, 64-way parallel across
// blocks, with A1[t+1] prefetch to hide L2 latency on the recurrence chain.
// ---------------------------------------------------------------------------

typedef __attribute__((ext_vector_type(2))) float v2f;
typedef __attribute__((ext_vector_type(8))) float v8f;

#define BATCH 64
#define TSTEPS 512
#define DIN 128
#define DHID 512
#define DOUT 128
#define NSLOT 8
#define SDIM 3
#define WRATE 0.1f

__device__ __forceinline__ float sigm(float x) {
    return 1.0f / (1.0f + expf(-x));
}

// ---------------------------------------------------------------------------
// Pack [K,3]x3 weight matrices into a zero-padded [K,16] matrix for WMMA.
// cols 0-2 = Wk, 3-5 = Ww, 6-8 = Wg, 9-15 = 0.
// ---------------------------------------------------------------------------
__global__ void pack3_kernel(const float* __restrict__ Wk,
                             const float* __restrict__ Ww,
                             const float* __restrict__ Wg,
                             float* __restrict__ Wpad, int K) {
    int i = blockIdx.x * blockDim.x + threadIdx.x;
    if (i >= K * 16) return;
    int r = i >> 4, j = i & 15;
    float v = 0.0f;
    if (j < 3)      v = Wk[r * 3 + j];
    else if (j < 6) v = Ww[r * 3 + j - 3];
    else if (j < 9) v = Wg[r * 3 + j - 6];
    Wpad[i] = v;
}

// ---------------------------------------------------------------------------
// Generic fp32 WMMA GEMM: C[M,N] = A[M,K] @ B[K,N]   (row major, lda=K, ldb=ldc=N)
// One wave per 16x16 C tile; V_WMMA_F32_16X16X4_F32, K stepped by 4.
// A-frag (16x4): lane l&15 = row; K = (vgpr + 2*(lane>>4)) + k  -> float2/lane.
// B-frag (4x16): K = (vgpr + 2*(lane>>4)) + k; N = lane&15.
// C/D (16x16): row = v + 8*(lane>>4); col = lane&15.
// ---------------------------------------------------------------------------
__global__ void wmma_gemm_f32(const float* __restrict__ A,
                              const float* __restrict__ B,
                              float* __restrict__ C,
                              int M, int N, int K) {
    int wave = blockIdx.x * (blockDim.x >> 5) + (threadIdx.x >> 5);
    int ntn = N >> 4;
    int total = (M >> 4) * ntn;
    if (wave >= total) return;          // wave-uniform exit: EXEC stays all-1s
    int tm = wave / ntn, tn = wave % ntn;
    int lane = threadIdx.x & 31;
    int half = lane >> 4;
    int l = lane & 15;

    const float* Arow = A + (size_t)(tm * 16 + l) * (size_t)K;
    const float* Bcol = B + tn * 16 + l;

    v8f c = {};
    for (int k = 0; k < K; k += 4) {
        v2f a;
        a.x = Arow[k + 2 * half];
        a.y = Arow[k + 2 * half + 1];
        v2f b;
        b.x = Bcol[(size_t)(k + 2 * half) * (size_t)N];
        b.y = Bcol[(size_t)(k + 2 * half + 1) * (size_t)N];
        c = __builtin_amdgcn_wmma_f32_16x16x4_f32(false, a, false, b,
                                                  (short)0, c, false, false);
    }
    float* Cp = C + (size_t)(tm * 16 + 8 * half) * (size_t)N + tn * 16 + l;
    #pragma unroll
    for (int v = 0; v < 8; ++v) Cp[(size_t)v * (size_t)N] = c[v];
}

// ---------------------------------------------------------------------------
// Scan 1 (layer 1). One block per batch element, 512 threads (one per Dhid).
// XO holds X1o = x @ W1o[:128] on entry; overwritten in place with
// o1 = sigmoid(X1o + read @ W1o[128:131] + b1o).
// Memory state [8*3] lives in LDS; k/v/g pre-activations (A1) prefetched.
// ---------------------------------------------------------------------------
__global__ void scan1_kernel(const float* __restrict__ A1,
                             const float* __restrict__ b1k,
                             const float* __restrict__ b1w,
                             const float* __restrict__ b1g,
                             float* __restrict__ XO,
                             const float* __restrict__ W1o,
                             const float* __restrict__ b1o,
                             float* __restrict__ mem_out) {
    const int b = blockIdx.x;
    const int tid = threadIdx.x;            // 0..511
    __shared__ float mem[NSLOT * SDIM];
    __shared__ float a1s[9];

    float bias9 = 0.0f;
    if (tid < 9)
        bias9 = (tid < 3) ? b1k[tid] : (tid < 6) ? b1w[tid - 3] : b1g[tid - 6];
    if (tid < NSLOT * SDIM) mem[tid] = 0.0f;
    if (tid < 9) a1s[tid] = A1[(size_t)(b * TSTEPS) * 16 + tid] + bias9;

    const float wr0 = W1o[(size_t)(DIN + 0) * DHID + tid];
    const float wr1 = W1o[(size_t)(DIN + 1) * DHID + tid];
    const float wr2 = W1o[(size_t)(DIN + 2) * DHID + tid];
    const float bo  = b1o[tid];
    __syncthreads();

    for (int t = 0; t < TSTEPS; ++t) {
        // prefetch next step's k/v/g pre-activations (hides L2 latency)
        float a1n = 0.0f;
        if (tid < 9 && t < TSTEPS - 1)
            a1n = A1[(size_t)(b * TSTEPS + t + 1) * 16 + tid] + bias9;
        float xo = XO[(size_t)(b * TSTEPS + t) * DHID + tid];

        float k0 = a1s[0], k1 = a1s[1], k2 = a1s[2];
        float v0 = a1s[3], v1 = a1s[4], v2 = a1s[5];
        float g0 = a1s[6], g1 = a1s[7], g2 = a1s[8];

        float sc[NSLOT];
        #pragma unroll
        for (int n = 0; n < NSLOT; ++n)
            sc[n] = mem[3 * n] * k0 + mem[3 * n + 1] * k1 + mem[3 * n + 2] * k2;
        float mx = sc[0];
        #pragma unroll
        for (int n = 1; n < NSLOT; ++n) mx = fmaxf(mx, sc[n]);
        float w[NSLOT], s = 0.0f;
        #pragma unroll
        for (int n = 0; n < NSLOT; ++n) { w[n] = expf(sc[n] - mx); s += w[n]; }
        float inv = 1.0f / s;
        float r0 = 0.0f, r1 = 0.0f, r2 = 0.0f;
        #pragma unroll
        for (int n = 0; n < NSLOT; ++n) {
            w[n] *= inv;
            r0 += w[n] * mem[3 * n];
            r1 += w[n] * mem[3 * n + 1];
            r2 += w[n] * mem[3 * n + 2];
        }
        float e0 = sigm(g0), e1 = sigm(g1), e2 = sigm(g2);

        float o = sigm(xo + r0 * wr0 + r1 * wr1 + r2 * wr2 + bo);
        XO[(size_t)(b * TSTEPS + t) * DHID + tid] = o;

        __syncthreads();                    // all reads of mem/a1s done
        if (tid < NSLOT * SDIM) {
            int n = tid / 3, d = tid - 3 * n;
            float wv = w[n];
            float ed = (d == 0) ? e0 : (d == 1) ? e1 : e2;
            float vd = (d == 0) ? v0 : (d == 1) ? v1 : v2;
            mem[tid] = mem[tid] * (1.0f - WRATE * wv * ed) + WRATE * wv * vd;
        }
        if (tid < 9 && t < TSTEPS - 1) a1s[tid] = a1n;
        __syncthreads();
    }
    if (tid < NSLOT * SDIM) mem_out[b * (NSLOT * SDIM) + tid] = mem[tid];
}

// ---------------------------------------------------------------------------
// Scan 2 (layer 2). State is tiny; one thread per batch element, everything in
// registers (fully unrolled constant indices). Records read_t at t = T-1.
// ---------------------------------------------------------------------------
__global__ void scan2_kernel(const float* __restrict__ A2,
                             const float* __restrict__ b2k,
                             const float* __restrict__ b2w,
                             const float* __restrict__ b2g,
                             const float* __restrict__ mem_in,
                             float* __restrict__ read_out) {
    int b = threadIdx.x;
    if (b >= BATCH) return;
    float mem[NSLOT * SDIM];
    #pragma unroll
    for (int i = 0; i < NSLOT * SDIM; ++i) mem[i] = mem_in[b * (NSLOT * SDIM) + i];
    float bk0 = b2k[0], bk1 = b2k[1], bk2 = b2k[2];
    float bw0 = b2w[0], bw1 = b2w[1], bw2 = b2w[2];
    float bg0 = b2g[0], bg1 = b2g[1], bg2 = b2g[2];
    float r0 = 0.0f, r1 = 0.0f, r2 = 0.0f;

    const float* Ab = A2 + (size_t)b * TSTEPS * 16;
    for (int t = 0; t < TSTEPS; ++t) {
        const float* a = Ab + (size_t)t * 16;
        float k0 = a[0] + bk0, k1 = a[1] + bk1, k2 = a[2] + bk2;
        float v0 = a[3] + bw0, v1 = a[4] + bw1, v2 = a[5] + bw2;
        float e0 = sigm(a[6] + bg0), e1 = sigm(a[7] + bg1), e2 = sigm(a[8] + bg2);

        float sc[NSLOT];
        #pragma unroll
        for (int n = 0; n < NSLOT; ++n)
            sc[n] = mem[3 * n] * k0 + mem[3 * n + 1] * k1 + mem[3 * n + 2] * k2;
        float mx = sc[0];
        #pragma unroll
        for (int n = 1; n < NSLOT; ++n) mx = fmaxf(mx, sc[n]);
        float w[NSLOT], s = 0.0f;
        #pragma unroll
        for (int n = 0; n < NSLOT; ++n) { w[n] = expf(sc[n] - mx); s += w[n]; }
        float inv = 1.0f / s;
        r0 = 0.0f; r1 = 0.0f; r2 = 0.0f;
        #pragma unroll
        for (int n = 0; n < NSLOT; ++n) {
            w[n] *= inv;
            r0 += w[n] * mem[3 * n];
            r1 += w[n] * mem[3 * n + 1];
            r2 += w[n] * mem[3 * n + 2];
        }
        #pragma unroll
        for (int n = 0; n < NSLOT; ++n) {
            float wv = w[n];
            mem[3 * n + 0] = mem[3 * n + 0] * (1.0f - WRATE * wv * e0) + WRATE * wv * v0;
            mem[3 * n + 1] = mem[3 * n + 1] * (1.0f - WRATE * wv * e1) + WRATE * wv * v1;
            mem[3 * n + 2] = mem[3 * n + 2] * (1.0f - WRATE * wv * e2) + WRATE * wv * v2;
        }
    }
    read_out[b * 3 + 0] = r0;   // read at t = T-1 (computed before final write)
    read_out[b * 3 + 1] = r1;
    read_out[b * 3 + 2] = r2;
}

// ---------------------------------------------------------------------------
// Final: out[64,128] = sigmoid(o1_last[64,512] @ W2o[:512] +
//                              read2[64,3] @ W2o[512:515] + b2o)
// o1_last row b lives at Alast + b*(512*512). WMMA fp32, epilogue fused.
// Exactly 32 waves = 4 blocks x 256 threads.
// ---------------------------------------------------------------------------
__global__ void wmma_final_kernel(const float* __restrict__ Alast,
                                  const float* __restrict__ W2o,
                                  const float* __restrict__ read2,
                                  const float* __restrict__ b2o,
                                  float* __restrict__ out) {
    int wave = blockIdx.x * (blockDim.x >> 5) + (threadIdx.x >> 5);
    int tm = wave >> 3;                 // 0..3  (M=64)
    int tn = wave & 7;                  // 0..7  (N=128)
    int lane = threadIdx.x & 31;
    int half = lane >> 4;
    int l = lane & 15;

    const size_t lda = (size_t)TSTEPS * DHID;   // stride between batch rows
    const float* Arow = Alast + (size_t)(tm * 16 + l) * lda;
    const float* Bcol = W2o + tn * 16 + l;

    v8f c = {};
    for (int k = 0; k < DHID; k += 4) {
        v2f a;
        a.x = Arow[k + 2 * half];
        a.y = Arow[k + 2 * half + 1];
        v2f bb;
        bb.x = Bcol[(size_t)(k + 2 * half) * DOUT];
        bb.y = Bcol[(size_t)(k + 2 * half + 1) * DOUT];
        c = __builtin_amdgcn_wmma_f32_16x16x4_f32(false, a, false, bb,
                                                  (short)0, c, false, false);
    }
    int col = tn * 16 + l;
    float w0 = W2o[(size_t)(DHID + 0) * DOUT + col];
    float w1 = W2o[(size_t)(DHID + 1) * DOUT + col];
    float w2 = W2o[(size_t)(DHID + 2) * DOUT + col];
    float bb2 = b2o[col];
    #pragma unroll
    for (int v = 0; v < 8; ++v) {
        int row = tm * 16 + 8 * half + v;
        float acc = c[v] + read2[row * 3 + 0] * w0 + read2[row * 3 + 1] * w1 +
                    read2[row * 3 + 2] * w2 + bb2;
        out[row * DOUT + col] = 1.0f / (1.0f + expf(-acc));
    }
}

// ---------------------------------------------------------------------------
extern "C" void kernel_launch(void* const* d_in, const int* in_sizes, int n_in,
                              void* d_out, int out_size, void* d_ws, size_t ws_size,
                              hipStream_t stream) {
    const float* inp = (const float*)d_in[0];
    const float* W1k = (const float*)d_in[1];
    const float* b1k = (const float*)d_in[2];
    const float* W1w = (const float*)d_in[3];
    const float* b1w = (const float*)d_in[4];
    const float* W1g = (const float*)d_in[5];
    const float* b1g = (const float*)d_in[6];
    const float* W1o = (const float*)d_in[7];
    const float* b1o = (const float*)d_in[8];
    const float* W2k = (const float*)d_in[9];
    const float* b2k = (const float*)d_in[10];
    const float* W2w = (const float*)d_in[11];
    const float* b2w = (const float*)d_in[12];
    const float* W2g = (const float*)d_in[13];
    const float* b2g = (const float*)d_in[14];
    const float* W2o = (const float*)d_in[15];
    const float* b2o = (const float*)d_in[16];
    float* out = (float*)d_out;

    const int BT = BATCH * TSTEPS;              // 32768 rows
    // Workspace layout (floats). XO is written as X1o then overwritten in
    // place with o1 (each element read once then rewritten by the same thread).
    float* ws   = (float*)d_ws;
    float* XO   = ws;                                   // BT*512 = 16,777,216
    float* A1   = XO + (size_t)BT * DHID;               // BT*16
    float* A2   = A1 + (size_t)BT * 16;                 // BT*16
    float* W1p  = A2 + (size_t)BT * 16;                 // 128*16
    float* W2p  = W1p + (size_t)DIN * 16;               // 512*16
    float* MEMF = W2p + (size_t)DHID * 16;              // 64*24
    float* RD2  = MEMF + (size_t)BATCH * NSLOT * SDIM;  // 64*3

    // 1) Pack the three [K,3] projection matrices into zero-padded [K,16].
    pack3_kernel<<<(DIN * 16 + 255) / 256, 256, 0, stream>>>(W1k, W1w, W1g, W1p, DIN);
    pack3_kernel<<<(DHID * 16 + 255) / 256, 256, 0, stream>>>(W2k, W2w, W2g, W2p, DHID);

    // 2) A1 = x @ W1pad            [32768 x 16 x 128]
    {
        int waves = (BT / 16) * (16 / 16);
        wmma_gemm_f32<<<(waves + 7) / 8, 256, 0, stream>>>(inp, W1p, A1, BT, 16, DIN);
    }
    // 3) X1o = x @ W1o[:128,:]     [32768 x 512 x 128]  (dominant GEMM, 4.3 GF)
    {
        int waves = (BT / 16) * (DHID / 16);
        wmma_gemm_f32<<<(waves + 7) / 8, 256, 0, stream>>>(inp, W1o, XO, BT, DHID, DIN);
    }
    // 4) Layer-1 scan: 64 blocks (one per batch), fused sigmoid, in-place o1.
    scan1_kernel<<<BATCH, DHID, 0, stream>>>(A1, b1k, b1w, b1g, XO, W1o, b1o, MEMF);

    // 5) A2 = o1 @ W2pad           [32768 x 16 x 512]
    {
        int waves = (BT / 16) * (16 / 16);
        wmma_gemm_f32<<<(waves + 7) / 8, 256, 0, stream>>>(XO, W2p, A2, BT, 16, DHID);
    }
    // 6) Layer-2 scan: tiny state, one thread per batch element.
    scan2_kernel<<<1, BATCH, 0, stream>>>(A2, b2k, b2w, b2g, MEMF, RD2);

    // 7) Final projection on last timestep only + read2 term + bias + sigmoid.
    wmma_final_kernel<<<4, 256, 0, stream>>>(XO + (size_t)(TSTEPS - 1) * DHID,
                                             W2o, RD2, b2o, out);
}